// MSMamba_86328842650188
// MI455X (gfx1250) — compile-verified
//
#include <hip/hip_runtime.h>
#include <hip/hip_bf16.h>

#define Nn    65536
#define CIN_  144
#define DIM_  128
#define NCLS_ 16
#define S1_   1024
#define S2_   512
#define DI_   256   // d_inner
#define DS_   16    // d_state

typedef __attribute__((ext_vector_type(2))) float v2f;
typedef __attribute__((ext_vector_type(8))) float v8f;

// ---------------------------------------------------------------------------
// WMMA helpers: D(16x16,f32) += A(16x4,f32) x B(4x16,f32), looped over K.
// A lives in LDS row-major [16 x lda]; B[k][n] = W[n*K + k] (weights row-major
// [ncols x K], i.e. the usual torch Linear weight, used as W^T on the right).
// Per ISA: A frag lane l holds A[l%16][ (l/16)*2 + j ], j=0..1. B mirrors.
// C/D frag: VGPR v holds row v + 8*(l/16), col l%16.
// ---------------------------------------------------------------------------
__device__ __forceinline__ v8f wmma_k4(v8f acc, float a0, float a1, float b0, float b1) {
  v2f a; a[0] = a0; a[1] = a1;
  v2f b; b[0] = b0; b[1] = b1;
  return __builtin_amdgcn_wmma_f32_16x16x4_f32(false, a, false, b, (short)0, acc,
                                               false, false);
}

__device__ __forceinline__ v8f gemm_tile(const float* As, int lda, int K,
                                         const float* __restrict__ W, int n, int lane) {
  v8f acc = {0.f, 0.f, 0.f, 0.f, 0.f, 0.f, 0.f, 0.f};
  const int m  = lane & 15;
  const int kh = (lane >> 4) << 1;
  const float* wrow = W + (size_t)n * K;
  for (int k = 0; k < K; k += 4) {
    acc = wmma_k4(acc, As[m * lda + k + kh], As[m * lda + k + kh + 1],
                  wrow[k + kh], wrow[k + kh + 1]);
  }
  return acc;
}

__device__ __forceinline__ float leakyf(float t) { return (t >= 0.f) ? t : 0.01f * t; }
__device__ __forceinline__ float siluf(float t)  { return t / (1.f + __expf(-t)); }

// ---------------------------------------------------------------------------
__global__ void k_zero(float* __restrict__ p, int nelem) {
  int i = blockIdx.x * 256 + threadIdx.x;
  if (i < nelem) p[i] = 0.f;
}

// denoise head: BN0 -> 1x1(144->128) -> leaky -> BN1 -> 1x1(128->128) -> leaky
__global__ void __launch_bounds__(256)
k_dn(const float* __restrict__ x,
     const float* g0, const float* b0, const float* m0, const float* v0,
     const float* __restrict__ w0, const float* bias0,
     const float* g1, const float* b1, const float* m1, const float* v1,
     const float* __restrict__ w1, const float* bias1,
     float* __restrict__ clean) {
  __shared__ float As[16 * CIN_];
  __shared__ float Ts[16 * DIM_];
  const int row0 = blockIdx.x * 16;
  const int tid  = threadIdx.x;
  for (int i = tid; i < 16 * CIN_; i += 256) {
    int r = i / CIN_, c = i - r * CIN_;
    float sc = g0[c] * __frsqrt_rn(v0[c] + 1e-5f);
    As[i] = (x[(size_t)(row0 + r) * CIN_ + c] - m0[c]) * sc + b0[c];
  }
  __syncthreads();
  const int lane = tid & 31;
  const int n    = (tid >> 5) * 16 + (lane & 15);
  v8f acc = gemm_tile(As, CIN_, CIN_, w0, n, lane);
  {
    float sc = g1[n] * __frsqrt_rn(v1[n] + 1e-5f);
    float bb = bias0[n];
    for (int v = 0; v < 8; ++v) {
      int m = v + ((lane >> 4) << 3);
      float t = leakyf(acc[v] + bb);
      Ts[m * DIM_ + n] = (t - m1[n]) * sc + b1[n];
    }
  }
  __syncthreads();
  v8f acc2 = gemm_tile(Ts, DIM_, DIM_, w1, n, lane);
  float bb = bias1[n];
  for (int v = 0; v < 8; ++v) {
    int m = v + ((lane >> 4) << 3);
    clean[(size_t)(row0 + m) * DIM_ + n] = leakyf(acc2[v] + bb);
  }
}

// SSConv point stage: BN -> 1x1(128->128, no bias) -> leaky
__global__ void __launch_bounds__(256)
k_bnpoint(const float* __restrict__ in,
          const float* g, const float* b, const float* mmn, const float* vvn,
          const float* __restrict__ w, float* __restrict__ out) {
  __shared__ float As[16 * DIM_];
  const int row0 = blockIdx.x * 16;
  const int tid  = threadIdx.x;
  for (int i = tid; i < 16 * DIM_; i += 256) {
    int r = i >> 7, c = i & 127;
    float sc = g[c] * __frsqrt_rn(vvn[c] + 1e-5f);
    As[i] = (in[(size_t)(row0 + r) * DIM_ + c] - mmn[c]) * sc + b[c];
  }
  __syncthreads();
  const int lane = tid & 31;
  const int n    = (tid >> 5) * 16 + (lane & 15);
  v8f acc = gemm_tile(As, DIM_, DIM_, w, n, lane);
  for (int v = 0; v < 8; ++v) {
    int m = v + ((lane >> 4) << 3);
    out[(size_t)(row0 + m) * DIM_ + n] = leakyf(acc[v]);
  }
}

// depthwise 5x5 (pad 2) + bias + leaky; thread = (pixel, 4 channels)
__global__ void __launch_bounds__(256)
k_dw5(const float* __restrict__ in, const float* __restrict__ wd,
      const float* __restrict__ bd, float* __restrict__ out) {
  int idx = blockIdx.x * 256 + threadIdx.x;        // Nn * 32
  int n  = idx >> 5;
  int cg = (idx & 31) << 2;
  int y  = n >> 8, xx = n & 255;
  float a0 = bd[cg], a1 = bd[cg + 1], a2 = bd[cg + 2], a3 = bd[cg + 3];
  for (int dy = -2; dy <= 2; ++dy) {
    int yy = y + dy; if (yy < 0 || yy > 255) continue;
    for (int dx = -2; dx <= 2; ++dx) {
      int xr = xx + dx; if (xr < 0 || xr > 255) continue;
      const float4 vv = *(const float4*)(in + ((size_t)((yy << 8) | xr)) * DIM_ + cg);
      int kk = (dy + 2) * 5 + (dx + 2);
      a0 += vv.x * wd[(cg + 0) * 25 + kk];
      a1 += vv.y * wd[(cg + 1) * 25 + kk];
      a2 += vv.z * wd[(cg + 2) * 25 + kk];
      a3 += vv.w * wd[(cg + 3) * 25 + kk];
    }
  }
  float4 r; r.x = leakyf(a0); r.y = leakyf(a1); r.z = leakyf(a2); r.w = leakyf(a3);
  *(float4*)(out + (size_t)n * DIM_ + cg) = r;
}

// extract one-hot segment index per pixel + per-segment counts
__global__ void k_seg(const float* __restrict__ Q, int S, int* __restrict__ seg,
                      float* __restrict__ cnt) {
  int n = blockIdx.x * 256 + threadIdx.x;
  if (n >= Nn) return;
  const float* row = Q + (size_t)n * S;
  int s = 0;
  for (int j = 0; j < S; ++j) { if (row[j] > 0.5f) { s = j; break; } }
  seg[n] = s;
  atomicAdd(&cnt[s], 1.0f);
}

// segsum[s][0:256] += [clean | cnn][n]
__global__ void k_segsum(const int* __restrict__ seg, const float* __restrict__ clean,
                         const float* __restrict__ cnn, float* __restrict__ ssum) {
  int idx = blockIdx.x * 256 + threadIdx.x;        // Nn * 64
  int n = idx >> 6;
  int f = (idx & 63) << 2;
  int s = seg[n];
  const float* src = (f < DIM_) ? (clean + (size_t)n * DIM_ + f)
                                : (cnn + (size_t)n * DIM_ + (f - DIM_));
  float4 v = *(const float4*)src;
  float* dst = ssum + (size_t)s * 256 + f;
  atomicAdd(dst + 0, v.x); atomicAdd(dst + 1, v.y);
  atomicAdd(dst + 2, v.z); atomicAdd(dst + 3, v.w);
}

// H2 = (segsum/count) @ lin1_w^T + lin1_b   [S,256]x[256,128]
__global__ void __launch_bounds__(256)
k_h2(const float* __restrict__ ssum, const float* __restrict__ cnt,
     const float* __restrict__ w, const float* __restrict__ b, float* __restrict__ H2) {
  __shared__ float As[16 * 256];
  const int s0 = blockIdx.x * 16, tid = threadIdx.x;
  for (int i = tid; i < 16 * 256; i += 256) {
    int r = i >> 8, c = i & 255;
    As[i] = ssum[(size_t)(s0 + r) * 256 + c] / fmaxf(cnt[s0 + r], 1e-9f);
  }
  __syncthreads();
  const int lane = tid & 31;
  const int n    = (tid >> 5) * 16 + (lane & 15);
  v8f acc = gemm_tile(As, 256, 256, w, n, lane);
  for (int v = 0; v < 8; ++v) {
    int m = v + ((lane >> 4) << 3);
    H2[(size_t)(s0 + m) * DIM_ + n] = acc[v] + b[n];
  }
}

// gather rows by order + in_proj: [L,128]x[128,512] -> xa0 [L,256] | z [L,256]
__global__ void __launch_bounds__(256)
k_inproj(const float* __restrict__ H2, const int* __restrict__ order,
         const float* __restrict__ w, float* __restrict__ xa0, float* __restrict__ z) {
  __shared__ float As[16 * DIM_];
  const int r0 = blockIdx.x * 16, tid = threadIdx.x;
  for (int i = tid; i < 16 * DIM_; i += 256) {
    int r = i >> 7, c = i & 127;
    As[i] = H2[(size_t)order[r0 + r] * DIM_ + c];
  }
  __syncthreads();
  const int lane = tid & 31;
  const int n    = blockIdx.y * 128 + (tid >> 5) * 16 + (lane & 15);
  v8f acc = gemm_tile(As, DIM_, DIM_, w, n, lane);
  for (int v = 0; v < 8; ++v) {
    int m = v + ((lane >> 4) << 3);
    int l = r0 + m;
    if (n < DI_) xa0[(size_t)l * DI_ + n] = acc[v];
    else         z[(size_t)l * DI_ + (n - DI_)] = acc[v];
  }
}

// causal depthwise conv1d (k=4) + bias + silu
__global__ void k_convsilu(const float* __restrict__ xa0, const float* __restrict__ cw,
                           const float* __restrict__ cb, float* __restrict__ xa, int L) {
  int idx = blockIdx.x * 256 + threadIdx.x;        // L * 256
  int l = idx >> 8, d = idx & 255;
  if (l >= L) return;
  float acc = cb[d];
#pragma unroll
  for (int j = 0; j < 4; ++j) {
    int t = l - 3 + j;
    if (t >= 0) acc += xa0[(size_t)t * DI_ + d] * cw[d * 4 + j];
  }
  xa[(size_t)l * DI_ + d] = siluf(acc);
}

// xdb = xa @ x_proj_w^T   [L,256]x[256,40]
__global__ void k_xproj(const float* __restrict__ xa, const float* __restrict__ w,
                        float* __restrict__ xdb, int L) {
  int idx = blockIdx.x * 256 + threadIdx.x;
  if (idx >= L * 40) return;
  int l = idx / 40, j = idx - l * 40;
  const float* a  = xa + (size_t)l * DI_;
  const float* wr = w + (size_t)j * DI_;
  float acc = 0.f;
  for (int k = 0; k < DI_; ++k) acc += a[k] * wr[k];
  xdb[(size_t)l * 40 + j] = acc;
}

// dt = softplus(dt_small @ dt_proj^T + b); dA = exp(dt*(-exp(Alog))); dBu = dt*xa*B
__global__ void k_dtdadbu(const float* __restrict__ xdb, const float* __restrict__ dtw,
                          const float* __restrict__ dtb, const float* __restrict__ Alog,
                          const float* __restrict__ xa, float* __restrict__ dA,
                          float* __restrict__ dBu, int L) {
  int idx = blockIdx.x * 256 + threadIdx.x;        // L * 256
  int l = idx >> 8, d = idx & 255;
  if (l >= L) return;
  const float* xr = xdb + (size_t)l * 40;
  float dt = dtb[d];
#pragma unroll
  for (int r = 0; r < 8; ++r) dt += xr[r] * dtw[d * 8 + r];
  dt = (dt > 20.f) ? dt : log1pf(__expf(dt));
  float xad = xa[(size_t)l * DI_ + d];
  float* pa = dA + ((size_t)l * DI_ + d) * DS_;
  float* pb = dBu + ((size_t)l * DI_ + d) * DS_;
#pragma unroll
  for (int s = 0; s < DS_; ++s) {
    float A = -__expf(Alog[d * DS_ + s]);
    pa[s] = __expf(dt * A);
    pb[s] = dt * xad * xr[8 + s];
  }
}

// selective scan: h = dA*h + dBu, y = <h,C>; fused +xa*D and *silu(z)
__global__ void __launch_bounds__(256)
k_scan(const float* __restrict__ dA, const float* __restrict__ dBu,
       const float* __restrict__ xdb, const float* __restrict__ xa,
       const float* __restrict__ z, const float* __restrict__ Dp,
       float* __restrict__ yfin, int L) {
  int d = threadIdx.x;
  float h[DS_];
#pragma unroll
  for (int s = 0; s < DS_; ++s) h[s] = 0.f;
  float Dd = Dp[d];
  for (int t = 0; t < L; ++t) {
    const float4* pa = (const float4*)(dA + ((size_t)t * DI_ + d) * DS_);
    const float4* pb = (const float4*)(dBu + ((size_t)t * DI_ + d) * DS_);
    const float* C = xdb + (size_t)t * 40 + 24;
    float dot = 0.f;
#pragma unroll
    for (int q = 0; q < 4; ++q) {
      float4 a = pa[q], b = pb[q];
      h[4 * q + 0] = a.x * h[4 * q + 0] + b.x;
      h[4 * q + 1] = a.y * h[4 * q + 1] + b.y;
      h[4 * q + 2] = a.z * h[4 * q + 2] + b.z;
      h[4 * q + 3] = a.w * h[4 * q + 3] + b.w;
      dot += h[4 * q + 0] * C[4 * q + 0] + h[4 * q + 1] * C[4 * q + 1]
           + h[4 * q + 2] * C[4 * q + 2] + h[4 * q + 3] * C[4 * q + 3];
    }
    float sz = siluf(z[(size_t)t * DI_ + d]);
    yfin[(size_t)t * DI_ + d] = (dot + xa[(size_t)t * DI_ + d] * Dd) * sz;
  }
}

// xg = yfin @ out_proj^T  [L,256]x[256,128], accumulating GroupNorm stats
__global__ void __launch_bounds__(256)
k_outproj(const float* __restrict__ yfin, const float* __restrict__ w,
          float* __restrict__ xg, float* __restrict__ stats) {
  __shared__ float As[16 * 256];
  __shared__ float rs[256], rq[256];
  const int r0 = blockIdx.x * 16, tid = threadIdx.x;
  for (int i = tid; i < 16 * 256; i += 256)
    As[i] = yfin[(size_t)r0 * 256 + i];
  __syncthreads();
  const int lane = tid & 31;
  const int n    = (tid >> 5) * 16 + (lane & 15);
  v8f acc = gemm_tile(As, 256, 256, w, n, lane);
  float lsum = 0.f, lss = 0.f;
  for (int v = 0; v < 8; ++v) {
    int m = v + ((lane >> 4) << 3);
    float val = acc[v];
    xg[(size_t)(r0 + m) * DIM_ + n] = val;
    lsum += val; lss += val * val;
  }
  rs[tid] = lsum; rq[tid] = lss;
  __syncthreads();
  for (int o = 128; o > 0; o >>= 1) {
    if (tid < o) { rs[tid] += rs[tid + o]; rq[tid] += rq[tid + o]; }
    __syncthreads();
  }
  if (tid == 0) { atomicAdd(&stats[0], rs[0]); atomicAdd(&stats[1], rq[0]); }
}

// GroupNorm(1,C) -> silu -> +residual(H2), reorder by rev, weight by softmax(weights1)
__global__ void k_post(const float* __restrict__ xg, const float* __restrict__ stats,
                       const float* __restrict__ gg, const float* __restrict__ gb,
                       const int* __restrict__ rev, const float* __restrict__ H2,
                       const float* __restrict__ w1p, int oidx, float* __restrict__ Gq,
                       int L) {
  int idx = blockIdx.x * 256 + threadIdx.x;        // L * 128
  int s = idx >> 7, d = idx & 127;
  if (s >= L) return;
  float cntf = (float)(L * DIM_);
  float mean = stats[0] / cntf;
  float var  = stats[1] / cntf - mean * mean;
  float inv  = __frsqrt_rn(var + 1e-5f);
  int l = rev[s];
  float v = (xg[(size_t)l * DIM_ + d] - mean) * inv * gg[d] + gb[d];
  v = siluf(v) + H2[(size_t)s * DIM_ + d];
  float a = w1p[d], b = w1p[DIM_ + d];
  float mx = fmaxf(a, b);
  float e0 = __expf(a - mx), e1 = __expf(b - mx);
  float wsel = ((oidx == 0) ? e0 : e1) / (e0 + e1);
  if (oidx == 0) Gq[(size_t)s * DIM_ + d]  = wsel * v;
  else           Gq[(size_t)s * DIM_ + d] += wsel * v;
}

// per-pixel gather + top-level weighting, write y1 = [G | cnn]
__global__ void k_gy1(const int* __restrict__ seg1, const int* __restrict__ seg2,
                      const float* __restrict__ G1, const float* __restrict__ G2,
                      const float* __restrict__ wp, const float* __restrict__ cnn,
                      float* __restrict__ y1) {
  int idx = blockIdx.x * 256 + threadIdx.x;        // Nn * 256
  int n = idx >> 8, f = idx & 255;
  float val;
  if (f < DIM_) {
    int d = f;
    float a = wp[d], b = wp[DIM_ + d];
    float mx = fmaxf(a, b);
    float e0 = __expf(a - mx), e1 = __expf(b - mx);
    float w0 = e0 / (e0 + e1);
    val = G1[(size_t)seg1[n] * DIM_ + d] * w0 + G2[(size_t)seg2[n] * DIM_ + d] * (1.f - w0);
  } else {
    val = cnn[(size_t)n * DIM_ + (f - DIM_)];
  }
  y1[(size_t)n * 256 + f] = val;
}

// classifier + softmax + 1e-29
__global__ void __launch_bounds__(256)
k_cls(const float* __restrict__ y1, const float* __restrict__ cw,
      const float* __restrict__ cb, float* __restrict__ Y) {
  __shared__ float Ws[16 * 256];
  __shared__ float Bs[16];
  int tid = threadIdx.x;
  for (int i = tid; i < 16 * 256; i += 256) Ws[i] = cw[i];
  if (tid < 16) Bs[tid] = cb[tid];
  __syncthreads();
  int n = blockIdx.x * 256 + tid;
  const float* row = y1 + (size_t)n * 256;
  float acc[16];
#pragma unroll
  for (int c = 0; c < 16; ++c) acc[c] = Bs[c];
  for (int k = 0; k < 256; ++k) {
    float v = row[k];
#pragma unroll
    for (int c = 0; c < 16; ++c) acc[c] += v * Ws[c * 256 + k];
  }
  float mx = acc[0];
#pragma unroll
  for (int c = 1; c < 16; ++c) mx = fmaxf(mx, acc[c]);
  float sum = 0.f;
#pragma unroll
  for (int c = 0; c < 16; ++c) { acc[c] = __expf(acc[c] - mx); sum += acc[c]; }
  float inv = 1.f / sum;
  float* out = Y + (size_t)n * 16;
#pragma unroll
  for (int c = 0; c < 16; ++c) out[c] = acc[c] * inv + 1e-29f;
}

// ---------------------------------------------------------------------------
static void run_mamba(hipStream_t stream, const float* H2, int L,
                      const int* order, const int* rev, int oidx, float* Gq,
                      const float* in_proj_w, const float* conv1d_w, const float* conv1d_b,
                      const float* x_proj_w, const float* dt_proj_w, const float* dt_proj_b,
                      const float* A_log, const float* Dp, const float* out_proj_w,
                      const float* gn_g, const float* gn_b, const float* weights1,
                      float* xa0, float* zb, float* xab, float* xdbb,
                      float* dAb, float* dBub, float* yfin, float* xgb, float* stats) {
  k_inproj<<<dim3(L / 16, 4), 256, 0, stream>>>(H2, order, in_proj_w, xa0, zb);
  k_convsilu<<<L, 256, 0, stream>>>(xa0, conv1d_w, conv1d_b, xab, L);
  k_xproj<<<(L * 40 + 255) / 256, 256, 0, stream>>>(xab, x_proj_w, xdbb, L);
  k_dtdadbu<<<L, 256, 0, stream>>>(xdbb, dt_proj_w, dt_proj_b, A_log, xab, dAb, dBub, L);
  k_scan<<<1, 256, 0, stream>>>(dAb, dBub, xdbb, xab, zb, Dp, yfin, L);
  k_zero<<<1, 256, 0, stream>>>(stats, 2);
  k_outproj<<<L / 16, 256, 0, stream>>>(yfin, out_proj_w, xgb, stats);
  k_post<<<L / 2, 256, 0, stream>>>(xgb, stats, gn_g, gn_b, rev, H2, weights1, oidx, Gq, L);
}

extern "C" void kernel_launch(void* const* d_in, const int* in_sizes, int n_in,
                              void* d_out, int out_size, void* d_ws, size_t ws_size,
                              hipStream_t stream) {
  (void)in_sizes; (void)n_in; (void)out_size; (void)ws_size;
  const float* x       = (const float*)d_in[0];
  const float* Q       = (const float*)d_in[1];
  const float* Q1      = (const float*)d_in[2];
  const int* order1    = (const int*)d_in[3];
  const int* rev1      = (const int*)d_in[4];
  const int* order1r   = (const int*)d_in[5];
  const int* rev1r     = (const int*)d_in[6];
  const int* order2    = (const int*)d_in[7];
  const int* rev2      = (const int*)d_in[8];
  const int* order2r   = (const int*)d_in[9];
  const int* rev2r     = (const int*)d_in[10];
  const float* dn_bn0_g = (const float*)d_in[11];
  const float* dn_bn0_b = (const float*)d_in[12];
  const float* dn_bn0_m = (const float*)d_in[13];
  const float* dn_bn0_v = (const float*)d_in[14];
  const float* dn_bn1_g = (const float*)d_in[15];
  const float* dn_bn1_b = (const float*)d_in[16];
  const float* dn_bn1_m = (const float*)d_in[17];
  const float* dn_bn1_v = (const float*)d_in[18];
  const float* ss0_bn_g = (const float*)d_in[19];
  const float* ss0_bn_b = (const float*)d_in[20];
  const float* ss0_bn_m = (const float*)d_in[21];
  const float* ss0_bn_v = (const float*)d_in[22];
  const float* ss1_bn_g = (const float*)d_in[23];
  const float* ss1_bn_b = (const float*)d_in[24];
  const float* ss1_bn_m = (const float*)d_in[25];
  const float* ss1_bn_v = (const float*)d_in[26];
  const float* dn_conv0_w = (const float*)d_in[27];
  const float* dn_conv0_b = (const float*)d_in[28];
  const float* dn_conv1_w = (const float*)d_in[29];
  const float* dn_conv1_b = (const float*)d_in[30];
  const float* ss0_point_w = (const float*)d_in[31];
  const float* ss0_depth_w = (const float*)d_in[32];
  const float* ss0_depth_b = (const float*)d_in[33];
  const float* ss1_point_w = (const float*)d_in[34];
  const float* ss1_depth_w = (const float*)d_in[35];
  const float* ss1_depth_b = (const float*)d_in[36];
  const float* in_proj_w  = (const float*)d_in[37];
  const float* conv1d_w   = (const float*)d_in[38];
  const float* conv1d_b   = (const float*)d_in[39];
  const float* x_proj_w   = (const float*)d_in[40];
  const float* dt_proj_w  = (const float*)d_in[41];
  const float* dt_proj_b  = (const float*)d_in[42];
  const float* A_log      = (const float*)d_in[43];
  const float* Dvec       = (const float*)d_in[44];
  const float* out_proj_w = (const float*)d_in[45];
  const float* gn_g       = (const float*)d_in[46];
  const float* gn_b       = (const float*)d_in[47];
  const float* lin1_w     = (const float*)d_in[48];
  const float* lin1_b     = (const float*)d_in[49];
  const float* cls_w      = (const float*)d_in[50];
  const float* cls_b      = (const float*)d_in[51];
  const float* weights    = (const float*)d_in[52];
  const float* weights1   = (const float*)d_in[53];

  // ---- workspace carve (deterministic) ----
  char* wsb = (char*)d_ws;
  size_t off = 0;
  auto carve = [&](size_t bytes) -> void* {
    void* p = wsb + off;
    off += (bytes + 255) & ~(size_t)255;
    return p;
  };
  float* clean = (float*)carve((size_t)Nn * DIM_ * 4);
  float* tp    = (float*)carve((size_t)Nn * DIM_ * 4);
  float* cnn   = (float*)carve((size_t)Nn * DIM_ * 4);
  int*   seg1  = (int*)carve((size_t)Nn * 4);
  int*   seg2  = (int*)carve((size_t)Nn * 4);
  float* cnt1  = (float*)carve(S1_ * 4);
  float* cnt2  = (float*)carve(S2_ * 4);
  float* ssum1 = (float*)carve((size_t)S1_ * 256 * 4);
  float* ssum2 = (float*)carve((size_t)S2_ * 256 * 4);
  float* H2a   = (float*)carve((size_t)S1_ * DIM_ * 4);
  float* H2b   = (float*)carve((size_t)S2_ * DIM_ * 4);
  float* G1    = (float*)carve((size_t)S1_ * DIM_ * 4);
  float* G2    = (float*)carve((size_t)S2_ * DIM_ * 4);
  float* xa0   = (float*)carve((size_t)S1_ * DI_ * 4);
  float* zb    = (float*)carve((size_t)S1_ * DI_ * 4);
  float* xab   = (float*)carve((size_t)S1_ * DI_ * 4);
  float* yfin  = (float*)carve((size_t)S1_ * DI_ * 4);
  float* xdbb  = (float*)carve((size_t)S1_ * 40 * 4);
  float* dAb   = (float*)carve((size_t)S1_ * DI_ * DS_ * 4);
  float* dBub  = (float*)carve((size_t)S1_ * DI_ * DS_ * 4);
  float* xgb   = (float*)carve((size_t)S1_ * DIM_ * 4);
  float* stats = (float*)carve(2 * 4);

  // ---- zero accumulators ----
  k_zero<<<(S1_ + 255) / 256, 256, 0, stream>>>(cnt1, S1_);
  k_zero<<<(S2_ + 255) / 256, 256, 0, stream>>>(cnt2, S2_);
  k_zero<<<(S1_ * 256 + 255) / 256, 256, 0, stream>>>(ssum1, S1_ * 256);
  k_zero<<<(S2_ * 256 + 255) / 256, 256, 0, stream>>>(ssum2, S2_ * 256);

  // ---- denoise + SSConv x2 ----
  k_dn<<<Nn / 16, 256, 0, stream>>>(x,
      dn_bn0_g, dn_bn0_b, dn_bn0_m, dn_bn0_v, dn_conv0_w, dn_conv0_b,
      dn_bn1_g, dn_bn1_b, dn_bn1_m, dn_bn1_v, dn_conv1_w, dn_conv1_b, clean);
  k_bnpoint<<<Nn / 16, 256, 0, stream>>>(clean, ss0_bn_g, ss0_bn_b, ss0_bn_m, ss0_bn_v,
                                         ss0_point_w, tp);
  k_dw5<<<Nn * 32 / 256, 256, 0, stream>>>(tp, ss0_depth_w, ss0_depth_b, cnn);
  k_bnpoint<<<Nn / 16, 256, 0, stream>>>(cnn, ss1_bn_g, ss1_bn_b, ss1_bn_m, ss1_bn_v,
                                         ss1_point_w, tp);
  k_dw5<<<Nn * 32 / 256, 256, 0, stream>>>(tp, ss1_depth_w, ss1_depth_b, cnn);

  // ---- superpixel segment means -> H2 (per Q level) ----
  k_seg<<<Nn / 256, 256, 0, stream>>>(Q, S1_, seg1, cnt1);
  k_seg<<<Nn / 256, 256, 0, stream>>>(Q1, S2_, seg2, cnt2);
  k_segsum<<<Nn * 64 / 256, 256, 0, stream>>>(seg1, clean, cnn, ssum1);
  k_segsum<<<Nn * 64 / 256, 256, 0, stream>>>(seg2, clean, cnn, ssum2);
  k_h2<<<S1_ / 16, 256, 0, stream>>>(ssum1, cnt1, lin1_w, lin1_b, H2a);
  k_h2<<<S2_ / 16, 256, 0, stream>>>(ssum2, cnt2, lin1_w, lin1_b, H2b);

  // ---- 4 Mamba runs (2 orders per Q level) ----
  run_mamba(stream, H2a, S1_, order1,  rev1,  0, G1, in_proj_w, conv1d_w, conv1d_b,
            x_proj_w, dt_proj_w, dt_proj_b, A_log, Dvec, out_proj_w, gn_g, gn_b,
            weights1, xa0, zb, xab, xdbb, dAb, dBub, yfin, xgb, stats);
  run_mamba(stream, H2a, S1_, order1r, rev1r, 1, G1, in_proj_w, conv1d_w, conv1d_b,
            x_proj_w, dt_proj_w, dt_proj_b, A_log, Dvec, out_proj_w, gn_g, gn_b,
            weights1, xa0, zb, xab, xdbb, dAb, dBub, yfin, xgb, stats);
  run_mamba(stream, H2b, S2_, order2,  rev2,  0, G2, in_proj_w, conv1d_w, conv1d_b,
            x_proj_w, dt_proj_w, dt_proj_b, A_log, Dvec, out_proj_w, gn_g, gn_b,
            weights1, xa0, zb, xab, xdbb, dAb, dBub, yfin, xgb, stats);
  run_mamba(stream, H2b, S2_, order2r, rev2r, 1, G2, in_proj_w, conv1d_w, conv1d_b,
            x_proj_w, dt_proj_w, dt_proj_b, A_log, Dvec, out_proj_w, gn_g, gn_b,
            weights1, xa0, zb, xab, xdbb, dAb, dBub, yfin, xgb, stats);

  // ---- head: y1 = [G | cnn], Y = softmax(y1 @ cls^T + b) + 1e-29 ----
  float* Yout  = (float*)d_out;
  float* y1out = (float*)d_out + (size_t)Nn * NCLS_;
  k_gy1<<<Nn, 256, 0, stream>>>(seg1, seg2, G1, G2, weights, cnn, y1out);
  k_cls<<<Nn / 256, 256, 0, stream>>>(y1out, cls_w, cls_b, Yout);
}